// DCRNN_81320910782822
// MI455X (gfx1250) — compile-verified
//
#include <hip/hip_runtime.h>
#include <hip/hip_bf16.h>

// ---------------------------------------------------------------------------
// DCRNN on gfx1250.
// Constants from the reference: B=4, T=12, N=10000, D=2, E=320000, H=64,
// K=2 Chebyshev steps, P=1, NPRED=12.
//
// Restructuring (all exact, uses linearity of propagation/GEMM):
//   layer0:  h1 = relu( x_t@W0x0 + px_t@W0x1 + h@W0h0 + ph@W0h1 + b0 )
//            where px_t = prop(x_t)   (precomputed for all t, x-only)
//                  ph   = prop(h)
//   layer1:  h  = relu( h1@W1a + prop(h1)@W1b + b1 )
//            W1a = W1[0][:64]+W1[0][64:],  W1b = W1[1][:64]+W1[1][64:]
// Final:     out = h@Wp + bp, broadcast over NPRED.
// ---------------------------------------------------------------------------

#define Bc 4
#define Tc 12
#define Nc 10000
#define Dc 2
#define Ec 320000
#define Hc 64
#define Rc (Bc * Nc)          // 40000 rows
#define NPREDc 12

typedef __attribute__((ext_vector_type(2))) float v2f;
typedef __attribute__((ext_vector_type(8))) float v8f;

// ---------------------------------------------------------------------------
// Fold weights: W0 [2][66][64], W1 [2][128][64] -> small dense 64x64 mats.
// ---------------------------------------------------------------------------
__global__ void fold_weights_kernel(const float* __restrict__ W0,
                                    const float* __restrict__ W1,
                                    float* __restrict__ W0x0,
                                    float* __restrict__ W0x1,
                                    float* __restrict__ W0h0,
                                    float* __restrict__ W0h1,
                                    float* __restrict__ W1a,
                                    float* __restrict__ W1b) {
  int i = blockIdx.x * blockDim.x + threadIdx.x;   // 0..4095
  if (i >= Hc * Hc) return;
  int c = i >> 6, o = i & 63;
  W0h0[i] = W0[(0 * 66 + (2 + c)) * 64 + o];
  W0h1[i] = W0[(1 * 66 + (2 + c)) * 64 + o];
  W1a[i] = W1[(0 * 128 + c) * 64 + o] + W1[(0 * 128 + 64 + c) * 64 + o];
  W1b[i] = W1[(1 * 128 + c) * 64 + o] + W1[(1 * 128 + 64 + c) * 64 + o];
  if (i < 2 * 64) {
    W0x0[i] = W0[(0 * 66) * 64 + i];   // rows 0..1 of slab 0
    W0x1[i] = W0[(1 * 66) * 64 + i];   // rows 0..1 of slab 1
  }
}

// ---------------------------------------------------------------------------
// One-time: px[t,b,n,0:2] = scatter-add over edges of ew[e] * x[b,t,src,0:2]
// ---------------------------------------------------------------------------
__global__ void scatter_x_kernel(const float* __restrict__ x,
                                 const int* __restrict__ src,
                                 const int* __restrict__ dst,
                                 const float* __restrict__ ew,
                                 float* __restrict__ px) {
  long long gid = (long long)blockIdx.x * blockDim.x + threadIdx.x;
  if (gid >= (long long)Tc * Bc * Ec) return;
  int e = (int)(gid % Ec);
  int tb = (int)(gid / Ec);
  int b = tb % Bc;
  int t = tb / Bc;
  int s = src[e], d = dst[e];
  float w = ew[e];
  const float* xp = x + (((long long)b * Tc + t) * Nc + s) * Dc;
  float* pp = px + (((long long)t * Bc + b) * Nc + d) * Dc;
  atomicAdd(pp + 0, w * xp[0]);
  atomicAdd(pp + 1, w * xp[1]);
}

// ---------------------------------------------------------------------------
// Per-step propagation of a 64-channel field: ph[b,dst] += ew * hsrc[b,src].
// One thread per (b, e, group-of-4-channels); float4 gathers hit L2 (h ~10MB),
// scatter-adds resolve as global_atomic_add_f32 in L2.
// ---------------------------------------------------------------------------
__global__ void scatter_h_kernel(const float* __restrict__ hsrc,
                                 const int* __restrict__ src,
                                 const int* __restrict__ dst,
                                 const float* __restrict__ ew,
                                 float* __restrict__ ph) {
  long long gid = (long long)blockIdx.x * blockDim.x + threadIdx.x;
  if (gid >= (long long)Bc * Ec * 16) return;
  int c4 = (int)(gid & 15);
  long long be = gid >> 4;
  int e = (int)(be % Ec);
  int b = (int)(be / Ec);
  int s = src[e], d = dst[e];
  float w = ew[e];
  const float4 v = *((const float4*)(hsrc + ((long long)b * Nc + s) * Hc) + c4);
  float* pp = ph + ((long long)b * Nc + d) * Hc + c4 * 4;
  atomicAdd(pp + 0, w * v.x);
  atomicAdd(pp + 1, w * v.y);
  atomicAdd(pp + 2, w * v.z);
  atomicAdd(pp + 3, w * v.w);
}

// ---------------------------------------------------------------------------
// Fused dual GEMM: out = relu( A1@M1 + A2@M2 + bias (+ x-epilogue) )
//   A1, A2: [R x 64], M1, M2: [64 x 64] row-major, out: [R x 64]
//
// WMMA V_WMMA_F32_16X16X4_F32 (exact f32). Per wave: one 16x16 D tile,
// 2 x 16 k-steps. Block = 256 threads = 8 wave32 = 32 rows x 64 cols.
// Both weight matrices are staged in LDS once per block (32 KB of the WGP's
// 320 KB) so the global pipe is reserved for streaming A rows; B fragments
// come from ds_load.
// VGPR layouts per ISA 7.12.2:
//   A 16x4:  lane L -> m = L&15, vgpr i -> k = 2*(L>>4)+i
//   B 4x16:  lane L -> n = L&15, vgpr i -> k = 2*(L>>4)+i
//   C 16x16: lane L -> n = L&15, vgpr j -> m = j + 8*(L>>4)
// ---------------------------------------------------------------------------
__global__ __launch_bounds__(256)
void gemm2_fused_kernel(const float* __restrict__ A1,
                        const float* __restrict__ M1,
                        const float* __restrict__ A2,
                        const float* __restrict__ M2,
                        const float* __restrict__ bias,
                        const float* __restrict__ x,     // null => no epilogue
                        int t,
                        const float* __restrict__ px_t,  // [R x 2]
                        const float* __restrict__ W0x0,  // [2 x 64]
                        const float* __restrict__ W0x1,  // [2 x 64]
                        float* __restrict__ out) {
  __shared__ float sM1[Hc * Hc];
  __shared__ float sM2[Hc * Hc];

  // Cooperative stage of both 64x64 weight matrices: 256 threads x 16 floats
  // each per matrix, as float4 -> global_load_b128 + ds_store_b128.
  {
    const int base = threadIdx.x * 16;
#pragma unroll
    for (int i = 0; i < 16; i += 4) {
      *(float4*)&sM1[base + i] = *(const float4*)&M1[base + i];
      *(float4*)&sM2[base + i] = *(const float4*)&M2[base + i];
    }
  }
  __syncthreads();

  const int lane   = threadIdx.x & 31;
  const int wave   = threadIdx.x >> 5;        // 0..7
  const int laneLo = lane & 15;
  const int laneHi = lane >> 4;               // 0 or 1
  const int rowBase = (blockIdx.x * 2 + (wave >> 2)) * 16;
  const int colBase = (wave & 3) * 16;
  const int row = rowBase + laneLo;
  const int col = colBase + laneLo;

  v8f acc = {};
  const float* Ar1 = A1 + (long long)row * Hc;
  const float* Ar2 = A2 + (long long)row * Hc;

#pragma unroll
  for (int k = 0; k < Hc; k += 4) {
    const int kk = k + 2 * laneHi;
    v2f a, b;
    a.x = Ar1[kk];
    a.y = Ar1[kk + 1];
    b.x = sM1[kk * Hc + col];
    b.y = sM1[(kk + 1) * Hc + col];
    acc = __builtin_amdgcn_wmma_f32_16x16x4_f32(false, a, false, b,
                                                (short)0, acc, false, false);
  }
#pragma unroll
  for (int k = 0; k < Hc; k += 4) {
    const int kk = k + 2 * laneHi;
    v2f a, b;
    a.x = Ar2[kk];
    a.y = Ar2[kk + 1];
    b.x = sM2[kk * Hc + col];
    b.y = sM2[(kk + 1) * Hc + col];
    acc = __builtin_amdgcn_wmma_f32_16x16x4_f32(false, a, false, b,
                                                (short)0, acc, false, false);
  }

  const float bcol = bias[col];
  float wx0 = 0.f, wx1 = 0.f, wp0 = 0.f, wp1 = 0.f;
  if (x != nullptr) {                 // uniform branch (scalar arg)
    wx0 = W0x0[col];
    wx1 = W0x0[64 + col];
    wp0 = W0x1[col];
    wp1 = W0x1[64 + col];
  }

#pragma unroll
  for (int j = 0; j < 8; ++j) {
    const int r = rowBase + j + 8 * laneHi;
    float v = acc[j] + bcol;
    if (x != nullptr) {
      const int b_ = r / Nc, n_ = r % Nc;
      const float* xp = x + (((long long)b_ * Tc + t) * Nc + n_) * Dc;
      const float* pp = px_t + (long long)r * Dc;
      v += xp[0] * wx0 + xp[1] * wx1 + pp[0] * wp0 + pp[1] * wp1;
    }
    v = fmaxf(v, 0.0f);
    out[(long long)r * Hc + col] = v;
  }
}

// ---------------------------------------------------------------------------
// Final projection h@Wp + bp, broadcast over NPRED time slots.
// ---------------------------------------------------------------------------
__global__ void predict_kernel(const float* __restrict__ h,
                               const float* __restrict__ Wp,
                               const float* __restrict__ bp,
                               float* __restrict__ out) {
  int r = blockIdx.x * blockDim.x + threadIdx.x;
  if (r >= Rc) return;
  const float* hr = h + (long long)r * Hc;
  float s = bp[0];
#pragma unroll
  for (int c = 0; c < Hc; ++c) s += hr[c] * Wp[c];
  const int b = r / Nc, n = r % Nc;
#pragma unroll
  for (int tp = 0; tp < NPREDc; ++tp)
    out[((long long)b * NPREDc + tp) * Nc + n] = s;
}

// ---------------------------------------------------------------------------
extern "C" void kernel_launch(void* const* d_in, const int* in_sizes, int n_in,
                              void* d_out, int out_size, void* d_ws, size_t ws_size,
                              hipStream_t stream) {
  const float* x    = (const float*)d_in[0];   // [B,T,N,D]
  const int*   ei   = (const int*)d_in[1];     // [2,E]
  const float* ew   = (const float*)d_in[2];   // [E]
  const float* W0   = (const float*)d_in[3];   // [2,66,64]
  const float* b0   = (const float*)d_in[4];   // [64]
  const float* W1   = (const float*)d_in[5];   // [2,128,64]
  const float* b1   = (const float*)d_in[6];   // [64]
  const float* Wp   = (const float*)d_in[7];   // [64,1]
  const float* bp   = (const float*)d_in[8];   // [1]
  const int* src = ei;
  const int* dst = ei + Ec;

  // workspace layout (floats)
  float* ws   = (float*)d_ws;
  float* h    = ws;                       // R*64
  float* h1   = h  + (size_t)Rc * Hc;     // R*64
  float* ph   = h1 + (size_t)Rc * Hc;     // R*64
  float* px   = ph + (size_t)Rc * Hc;     // T*R*2
  float* W0x0 = px + (size_t)Tc * Rc * Dc;
  float* W0x1 = W0x0 + 2 * 64;
  float* W0h0 = W0x1 + 2 * 64;
  float* W0h1 = W0h0 + 64 * 64;
  float* W1a  = W0h1 + 64 * 64;
  float* W1b  = W1a  + 64 * 64;

  // h0 = 0, px = 0 (graph-capturable memset nodes)
  hipMemsetAsync(h,  0, (size_t)Rc * Hc * sizeof(float), stream);
  hipMemsetAsync(px, 0, (size_t)Tc * Rc * Dc * sizeof(float), stream);

  fold_weights_kernel<<<(Hc * Hc + 255) / 256, 256, 0, stream>>>(
      W0, W1, W0x0, W0x1, W0h0, W0h1, W1a, W1b);

  // prop(x_t) for all t in parallel (x-only, recurrence-independent)
  {
    long long total = (long long)Tc * Bc * Ec;
    scatter_x_kernel<<<(unsigned)((total + 255) / 256), 256, 0, stream>>>(
        x, src, dst, ew, px);
  }

  const unsigned scatterBlocks = (unsigned)(((long long)Bc * Ec * 16 + 255) / 256);
  const unsigned gemmBlocks = Rc / 32;   // 1250

  for (int t = 0; t < Tc; ++t) {
    // layer 0
    hipMemsetAsync(ph, 0, (size_t)Rc * Hc * sizeof(float), stream);
    scatter_h_kernel<<<scatterBlocks, 256, 0, stream>>>(h, src, dst, ew, ph);
    gemm2_fused_kernel<<<gemmBlocks, 256, 0, stream>>>(
        h, W0h0, ph, W0h1, b0, x, t, px + (size_t)t * Rc * Dc, W0x0, W0x1, h1);
    // layer 1
    hipMemsetAsync(ph, 0, (size_t)Rc * Hc * sizeof(float), stream);
    scatter_h_kernel<<<scatterBlocks, 256, 0, stream>>>(h1, src, dst, ew, ph);
    gemm2_fused_kernel<<<gemmBlocks, 256, 0, stream>>>(
        h1, W1a, ph, W1b, b1, nullptr, 0, nullptr, nullptr, nullptr, h);
  }

  predict_kernel<<<(Rc + 255) / 256, 256, 0, stream>>>(h, Wp, bp, (float*)d_out);
}